// KvvGaussianLayer_47450798686618
// MI455X (gfx1250) — compile-verified
//
#include <hip/hip_runtime.h>
#include <hip/hip_bf16.h>

typedef __attribute__((ext_vector_type(2))) float v2f;
typedef __attribute__((ext_vector_type(8))) float v8f;

#define TT 4096
#define NB 4
#define NF 131            // 1 + 128 + 1 + 1
#define DK 128            // embedding dim
#define LDSP 129          // padded LDS row stride (129 mod 64 == 1 -> conflict free)
#define ZSCALE 0.08838834764831845f  // 1/sqrt(128)

// ---------------------------------------------------------------------------
// Kernel 1: per-row scalars. mean -> d_out, sq / v / softplus(noise) -> d_ws
// ---------------------------------------------------------------------------
__global__ __launch_bounds__(256) void prep_kernel(const float* __restrict__ in,
                                                   float* __restrict__ mean_out,
                                                   float* __restrict__ sq_out,
                                                   float* __restrict__ v_out,
                                                   float* __restrict__ ns_out) {
    const int idx = blockIdx.x * 256 + threadIdx.x;   // 0 .. B*T-1
    if (idx >= NB * TT) return;
    const size_t base = (size_t)idx * NF;

    mean_out[idx] = in[base];

    float acc = 0.0f;
    #pragma unroll 8
    for (int k = 0; k < DK; ++k) {
        float z = in[base + 1 + k];
        acc += z * z;
    }
    sq_out[idx] = acc * (1.0f / (float)DK);   // sum of (z/sqrt(DK))^2

    v_out[idx] = in[base + 1 + DK];

    float x = in[base + 2 + DK];
    // numerically stable softplus
    float sp = (x > 0.0f) ? (x + log1pf(__expf(-x))) : log1pf(__expf(x));
    ns_out[idx] = sp;
}

// ---------------------------------------------------------------------------
// Kernel 2: covariance tiles via fp32 WMMA (V_WMMA_F32_16X16X4_F32).
// Block = 8 waves as 2 (i) x 4 (j); block tile = 32 x 64 of the T x T output.
// Streaming outputs use non-temporal stores (TH=NT) so the 537 MB of
// write-once data doesn't thrash L2 (which holds the reused 8 MB Z set).
// ---------------------------------------------------------------------------
__global__ __launch_bounds__(256) void cov_kernel(const float* __restrict__ in,
                                                  const float* __restrict__ sq,
                                                  const float* __restrict__ vv,
                                                  const float* __restrict__ ns,
                                                  float* __restrict__ fcov,
                                                  float* __restrict__ ycov) {
    __shared__ float As[32 * LDSP];   // 32 rows of Z (i-strip), pre-scaled
    __shared__ float Bs[64 * LDSP];   // 64 rows of Z (j-strip), pre-scaled

    const int tid   = threadIdx.x;
    const int bb    = blockIdx.z;
    const int ibase = blockIdx.y * 32;
    const int jbase = blockIdx.x * 64;

    // ---- stage A strip: 32 x 128 floats, coalesced b32 loads ----
    for (int k = tid; k < 32 * DK; k += 256) {
        int r = k >> 7, c = k & (DK - 1);
        As[r * LDSP + c] = ZSCALE * in[(size_t)(bb * TT + ibase + r) * NF + 1 + c];
    }
    // ---- stage B strip: 64 x 128 floats ----
    for (int k = tid; k < 64 * DK; k += 256) {
        int r = k >> 7, c = k & (DK - 1);
        Bs[r * LDSP + c] = ZSCALE * in[(size_t)(bb * TT + jbase + r) * NF + 1 + c];
    }
    __syncthreads();

    const int wave   = tid >> 5;
    const int lane   = tid & 31;
    const int wave_i = wave >> 2;        // 0..1
    const int wave_j = wave & 3;         // 0..3
    const int m      = lane & 15;        // row-in-tile selector
    const int h      = lane >> 4;        // lane half

    const int arow = wave_i * 16 + m;    // A row in LDS strip
    const int brow = wave_j * 16 + m;    // B row in LDS strip

    v8f acc = {};
    #pragma unroll
    for (int kk = 0; kk < DK; kk += 4) {
        // A fragment (16x4 f32): VGPR0 = K+0 (lanes 0-15) / K+2 (16-31),
        //                        VGPR1 = K+1 (lanes 0-15) / K+3 (16-31)
        v2f a;
        a.x = As[arow * LDSP + kk + 2 * h];
        a.y = As[arow * LDSP + kk + 2 * h + 1];
        // B fragment (4x16 f32): row K striped across lanes (N = m),
        //   VGPR0 = K+0 / K+1 per lane half, VGPR1 = K+2 / K+3
        v2f b;
        b.x = Bs[brow * LDSP + kk + h];
        b.y = Bs[brow * LDSP + kk + 2 + h];

        acc = __builtin_amdgcn_wmma_f32_16x16x4_f32(
            /*neg_a=*/false, a, /*neg_b=*/false, b,
            /*c_mod=*/(short)0, acc, /*reuse_a=*/false, /*reuse_b=*/false);
    }

    // ---- epilogue: C layout = VGPR r -> M = r + 8*h, lane -> N = m ----
    const int jg  = jbase + wave_j * 16 + m;
    const float sqj = sq[bb * TT + jg];
    const float vj  = vv[bb * TT + jg];

    #pragma unroll
    for (int r = 0; r < 8; ++r) {
        const int ig = ibase + wave_i * 16 + r + 8 * h;
        const float sqi = sq[bb * TT + ig];
        const float vi  = vv[bb * TT + ig];
        float val = __expf(acc[r] - 0.5f * (sqi + sqj)) * vi * vj;
        float f = val, y = val;
        if (ig == jg) {
            f += 1e-6f;
            y += 1e-6f + ns[bb * TT + ig];
        }
        const size_t off = ((size_t)bb * TT + (size_t)ig) * TT + (size_t)jg;
        __builtin_nontemporal_store(f, &fcov[off]);
        __builtin_nontemporal_store(y, &ycov[off]);
    }
}

// ---------------------------------------------------------------------------
extern "C" void kernel_launch(void* const* d_in, const int* in_sizes, int n_in,
                              void* d_out, int out_size, void* d_ws, size_t ws_size,
                              hipStream_t stream) {
    (void)in_sizes; (void)n_in; (void)out_size; (void)ws_size;

    const float* in = (const float*)d_in[0];
    float* out  = (float*)d_out;
    float* mean = out;                                   // 4*4096
    float* fcov = out + (size_t)NB * TT;                 // 4*4096*4096
    float* ycov = fcov + (size_t)NB * TT * TT;

    float* sq = (float*)d_ws;                            // 16384 floats
    float* vv = sq + NB * TT;
    float* ns = vv + NB * TT;

    prep_kernel<<<(NB * TT) / 256, 256, 0, stream>>>(in, mean, sq, vv, ns);

    dim3 grid(TT / 64, TT / 32, NB);                     // 64 x 128 x 4 blocks
    cov_kernel<<<grid, 256, 0, stream>>>(in, sq, vv, ns, fcov, ycov);
}